// StateAttention_1580547969247
// MI455X (gfx1250) — compile-verified
//
#include <hip/hip_runtime.h>
#include <hip/hip_bf16.h>
#include <stdint.h>

typedef float v2f __attribute__((ext_vector_type(2)));
typedef float v8f __attribute__((ext_vector_type(8)));

#define TT 4096
#define DD 512
#define TILE_D 128
#define NWAVES 8
#define ST 4                    // t-rows per staged chunk (one WMMA K-chunk)
#define NBUF 4                  // 4-deep staging: 3 chunks (6KB/wave) in flight
#define TPW (TT / NWAVES)       // 512 t per wave
#define NSTEP (TPW / ST)        // 128 steps

// ---------------- Kernel 1: new_a_t stencil ----------------
__global__ void __launch_bounds__(256)
new_at_kernel(const float* __restrict__ a, const float* __restrict__ r,
              const float* __restrict__ mask, float* __restrict__ new_a) {
  const int b = blockIdx.x;
  const float r0 = r[b * 2 + 0];
  const float r1 = r[b * 2 + 1];
  const float* ab = a + (size_t)b * TT;
  const float* mb = mask + (size_t)b * TT;
  float* ob = new_a + (size_t)b * TT;
  for (int t = threadIdx.x; t < TT; t += 256) {
    float v;
    if (t == 0) v = ab[0] * r0;                                   // a[0]*r0
    else        v = mb[t] * (ab[t - 1] * r1 + ab[t] * r0);        // m[t-1]*ws[t-1]
    if (t < TT - 1)
      v += (1.0f - mb[t + 1]) * (ab[t] * r1 + ab[t + 1] * r0);    // (1-m[t])*ws[t]
    ob[t] = v;
  }
}

// low 32 bits of a flat LDS pointer == LDS byte offset
__device__ __forceinline__ unsigned lds_off_u32(const void* p) {
  return (unsigned)(uintptr_t)p;
}

// ---------------- Kernel 2: streaming weighted reduction ----------------
__global__ void __launch_bounds__(256)
reduce_kernel(const float* __restrict__ E, const float* __restrict__ w_g,
              float* __restrict__ out) {
  __shared__ float w_s[TT];                          // 16 KB weights
  __shared__ float stage[NWAVES][NBUF][ST][TILE_D];  // 64 KB, 4-deep staging
  __shared__ float red[NWAVES][TILE_D];              // 4 KB cross-wave reduce

  const int b   = blockIdx.y;
  const int d0  = blockIdx.x * TILE_D;
  const int tid = threadIdx.x;
  const int wv  = tid >> 5;
  const int l   = tid & 31;

  for (int i = tid; i < TT; i += 256) w_s[i] = w_g[(size_t)b * TT + i];
  __syncthreads();

  const uint64_t Ebase = (uint64_t)(uintptr_t)(E + (size_t)b * TT * DD);
  const int tw   = wv * TPW;
  const int krow = (l >> 4) << 1;   // lanes 0-15 -> K{0,1}, lanes 16-31 -> K{2,3}
  const int col  = l & 15;

  v8f C[8] = {};                    // 8 d-subtiles of 16

  const unsigned ldsq = lds_off_u32(&stage[wv][0][0][l * 4]);

  // stage chunk (ST rows of 128 floats) into buffer `buf`:
  // one NT async b128 per lane per row (512B/instr), streamed past L2
  auto issue = [&](int buf, int chunk) {
    unsigned ldsd = ldsq + (unsigned)buf * (ST * TILE_D * 4);
    uint64_t g = Ebase +
        ((uint64_t)(tw + chunk * ST) * DD + (unsigned)(d0 + l * 4)) * 4ull;
#pragma unroll
    for (int rr = 0; rr < ST; ++rr) {
      asm volatile("global_load_async_to_lds_b128 %0, %1, off th:TH_LOAD_NT"
                   :: "v"(ldsd), "v"(g) : "memory");
      ldsd += TILE_D * 4;
      g    += (uint64_t)DD * 4;
    }
  };

  // consume one landed chunk with 8 broadcast-row WMMAs
  auto consume = [&](int step) {
    const int cur = step & (NBUF - 1);
    const int tb  = tw + step * ST;
    v2f A;                              // rows replicated -> per-lane w pair
    A[0] = w_s[tb + krow];
    A[1] = w_s[tb + krow + 1];
    const float* sbase = &stage[wv][cur][krow][col];
#pragma unroll
    for (int n = 0; n < 8; ++n) {
      v2f Bv;
      Bv[0] = sbase[n * 16];            // B[k,   d0+n*16+col]
      Bv[1] = sbase[n * 16 + TILE_D];   // B[k+1, d0+n*16+col]
      C[n] = __builtin_amdgcn_wmma_f32_16x16x4_f32(false, A, false, Bv,
                                                   (short)0, C[n], false, false);
    }
  };

  // prologue: fill 3 of 4 buffers
  issue(0, 0);
  issue(1, 1);
  issue(2, 2);

  // steady state: unconditional issue + wait + consume.
  // unroll(1): keep one WMMA set so accumulators stay tied in place
  int step = 0;
#pragma unroll 1
  for (; step + 3 < NSTEP; ++step) {
    issue((step + 3) & (NBUF - 1), step + 3);
    asm volatile("s_wait_asynccnt 12" ::: "memory");  // chunk `step` landed
    consume(step);
  }
  // epilogue: drain last 3 chunks
  asm volatile("s_wait_asynccnt 8" ::: "memory");
  consume(step++);
  asm volatile("s_wait_asynccnt 4" ::: "memory");
  consume(step++);
  asm volatile("s_wait_asynccnt 0" ::: "memory");
  consume(step);

  // every row of C holds the full sum; take M=0 (VGPR 0, lanes 0-15)
  if (l < 16) {
#pragma unroll
    for (int n = 0; n < 8; ++n) red[wv][n * 16 + l] = C[n][0];
  }
  __syncthreads();
  if (tid < TILE_D) {
    float s = 0.0f;
#pragma unroll
    for (int w2 = 0; w2 < NWAVES; ++w2) s += red[w2][tid];
    out[(size_t)b * DD + d0 + tid] = s;
  }
}

extern "C" void kernel_launch(void* const* d_in, const int* in_sizes, int n_in,
                              void* d_out, int out_size, void* d_ws, size_t ws_size,
                              hipStream_t stream) {
  const float* a_t  = (const float*)d_in[0];   // [64,4096]
  const float* r_t  = (const float*)d_in[1];   // [64,2]
  const float* E    = (const float*)d_in[2];   // [64,4096,512]
  const float* mask = (const float*)d_in[3];   // [64,4096]

  float* out   = (float*)d_out;                // output [64,512] first
  float* new_a = out + 64 * 512;               // then new_a_t [64,4096]

  new_at_kernel<<<64, 256, 0, stream>>>(a_t, r_t, mask, new_a);

  dim3 grid(DD / TILE_D, 64);                  // 4 x 64 = 256 blocks
  reduce_kernel<<<grid, 256, 0, stream>>>(E, new_a, out);
}